// SpikeAndSlabSampler_18734647345743
// MI455X (gfx1250) — compile-verified
//
#include <hip/hip_runtime.h>

// SpikeAndSlabSampler for MI455X (gfx1250, wave32).
// Outputs concatenated flat: z[K*P], z_mean[P], theta[K*P], alpha[P], logalpha[P].
//
// ZETA=1.1, GAMMA=-0.1, BETA=2/3, GZ_LOGRATIO=-GAMMA/ZETA=1/11
//   shift      = BETA*GZ_LOGRATIO = 2/33
//   1/BETA     = 1.5
//   ZETA-GAMMA = 1.2
//
// Gate identity used below:
//   s = sigmoid((log(u) - log1p(-u) + la)/BETA)
//     = 1 / (1 + exp(-1.5*la) * ((1-u)/u)^1.5)
//   ((1-u)/u)^1.5 = w*sqrt(w), w = (1-u)*rcp(u)
// -> per element: v_rcp + v_sqrt + v_rcp (3 trans ops) instead of
//    rcp + v_log + v_exp + rcp (4 trans ops); exp(-1.5*la) hoisted per column.

#define BLOCK 256

template <int KT>  // KT > 0: compile-time K (gate loop fully unrolled); KT == 0: runtime K
__global__ __launch_bounds__(BLOCK) void sss_kernel(
    const float* __restrict__ Z,
    const float* __restrict__ U,
    const float* __restrict__ u,
    const float* __restrict__ gamma_alpha,
    const float* __restrict__ logalpha,
    float* __restrict__ z_out,
    float* __restrict__ zmean_out,
    float* __restrict__ theta_out,
    float* __restrict__ alpha_out,
    float* __restrict__ la_out,
    int P, int B, int K)
{
    const int p = blockIdx.x * BLOCK + threadIdx.x;
    if (p >= P) return;

    const int Keff = (KT > 0) ? KT : K;
    const size_t sp = (size_t)p;
    const size_t Ps = (size_t)P;

    const float ga = gamma_alpha[p];
    const float la = logalpha[p];

    // --- Marsaglia-Tsang setup ---
    const float alpha = fmaxf(ga, 0.0f) + 1.0f;            // relu(ga) + 1
    const float d     = alpha - (1.0f / 3.0f);
    const float c     = __builtin_amdgcn_rsqf(9.0f * d);   // 1/sqrt(9d), v_rsq_f32

    // --- rejection scan: keep FIRST K accepted d*V per column; early exit ---
    int cnt = 0;
    for (int b = 0; b < B; ++b) {
        if (cnt >= Keff) break;                  // first-K semantics: later rows irrelevant
        const size_t off = (size_t)b * Ps + sp;
        if (b + 4 < B) {                         // global_prefetch_b8, latency hiding
            __builtin_prefetch(&Z[off + 4 * Ps], 0, 3);
            __builtin_prefetch(&U[off + 4 * Ps], 0, 3);
        }
        const float z  = __builtin_nontemporal_load(&Z[off]);   // streamed once: TH=NT
        const float Uv = __builtin_nontemporal_load(&U[off]);

        const float t  = fmaf(c, z, 1.0f);       // cond1: Z > -1/c  <=>  t > 0
        const float V  = t * t * t;
        const float dV = d * V;
        // cond2: log U < 0.5 z^2 + d - d V + d log V   (NaN compares false, as in JAX)
        const float rhs  = fmaf(0.5f * z, z, d) - dV + d * __logf(V);
        const bool  cond = (t > 0.0f) && (__logf(Uv) < rhs);
        if (cond) {
            __builtin_nontemporal_store(dV, &theta_out[(size_t)cnt * Ps + sp]);
            ++cnt;
        }
    }
    for (int k = cnt; k < Keff; ++k)             // zero-pad (matches out=0 for rejected)
        __builtin_nontemporal_store(0.0f, &theta_out[(size_t)k * Ps + sp]);

    // --- hard-concrete spike-and-slab gate (training mode) ---
    const float shift = 0.060606062f;            // BETA * (-GAMMA/ZETA) = 2/33
    const float zmg   = 1.2f;                    // ZETA - GAMMA
    const float gam   = -0.1f;                   // GAMMA

    {   // z_mean = sigmoid(la - shift)
        const float e = __expf(-(la - shift));
        __builtin_nontemporal_store(__builtin_amdgcn_rcpf(1.0f + e), &zmean_out[p]);
    }

    const float E = __expf(-1.5f * la);          // exp(-la/BETA), hoisted per column

    #pragma unroll
    for (int k = 0; k < Keff; ++k) {
        const float uu = __builtin_nontemporal_load(&u[(size_t)k * Ps + sp]);
        const float w  = (1.0f - uu) * __builtin_amdgcn_rcpf(uu);   // (1-u)/u
        const float g  = w * __builtin_amdgcn_sqrtf(w);             // w^1.5
        const float s  = __builtin_amdgcn_rcpf(fmaf(E, g, 1.0f));   // sigmoid
        const float zz = fmaf(zmg, s, gam);                         // 1.2*s - 0.1
        __builtin_nontemporal_store(fminf(fmaxf(zz, 0.0f), 1.0f),
                                    &z_out[(size_t)k * Ps + sp]);
    }

    __builtin_nontemporal_store(alpha, &alpha_out[p]);
    __builtin_nontemporal_store(la,    &la_out[p]);
}

extern "C" void kernel_launch(void* const* d_in, const int* in_sizes, int n_in,
                              void* d_out, int out_size, void* d_ws, size_t ws_size,
                              hipStream_t stream)
{
    const float* Z  = (const float*)d_in[0];   // [B, P]
    const float* U  = (const float*)d_in[1];   // [B, P]
    const float* u  = (const float*)d_in[2];   // [K, P]
    const float* ga = (const float*)d_in[3];   // [P]
    const float* la = (const float*)d_in[4];   // [P]

    const int P = in_sizes[3];
    const int B = in_sizes[0] / P;
    const int K = in_sizes[2] / P;

    float* out       = (float*)d_out;
    float* z_out     = out;                          // K*P
    float* zmean_out = z_out + (size_t)K * P;        // P
    float* theta_out = zmean_out + P;                // K*P
    float* alpha_out = theta_out + (size_t)K * P;    // P
    float* la_out    = alpha_out + P;                // P

    const int grid = (P + BLOCK - 1) / BLOCK;
    if (K == 10) {
        sss_kernel<10><<<grid, BLOCK, 0, stream>>>(Z, U, u, ga, la,
                                                   z_out, zmean_out, theta_out,
                                                   alpha_out, la_out, P, B, K);
    } else {
        sss_kernel<0><<<grid, BLOCK, 0, stream>>>(Z, U, u, ga, la,
                                                  z_out, zmean_out, theta_out,
                                                  alpha_out, la_out, P, B, K);
    }
}